// wFMLayer_51092930953753
// MI455X (gfx1250) — compile-verified
//
#include <hip/hip_runtime.h>

// Problem constants from the reference
#define B_   4
#define N_   4096
#define D_   4
#define C_   16
#define KNN  32      // top-k neighbors
#define M_   32
#define EPSF 1e-4f

typedef __attribute__((ext_vector_type(2))) float v2f;
typedef __attribute__((ext_vector_type(8))) float v8f;

// ---------------------------------------------------------------------------
// Kernel 1: normalize weights.
//   w1n[c][k] = w1[c][k]^2 / sum_k w1[c][k]^2                  (C x KNN)
//   w2n[c][m] = w2[m][c]^2 / sum_c w2[m][c]^2  (stored [C][M]) (C x M)
// ---------------------------------------------------------------------------
__global__ void wfm_normalize_weights(const float* __restrict__ w1,
                                      const float* __restrict__ w2,
                                      float* __restrict__ w1n,
                                      float* __restrict__ w2n) {
  int t = threadIdx.x;
  if (t < C_) {
    int c = t;
    float s = 0.f;
    for (int k = 0; k < KNN; ++k) { float v = w1[c * KNN + k]; s += v * v; }
    float inv = 1.0f / s;
    for (int k = 0; k < KNN; ++k) {
      float v = w1[c * KNN + k];
      w1n[c * KNN + k] = v * v * inv;
    }
  } else if (t < C_ + M_) {
    int m = t - C_;
    float s = 0.f;
    for (int c = 0; c < C_; ++c) { float v = w2[m * C_ + c]; s += v * v; }
    float inv = 1.0f / s;
    for (int c = 0; c < C_; ++c) {
      float v = w2[m * C_ + c];
      w2n[c * M_ + m] = v * v * inv;   // transposed store -> [C][M]
    }
  }
}

// ---------------------------------------------------------------------------
// Kernel 2: sorted top-32 per row of adj[B*N][N]  (the bandwidth-bound phase:
// 256 MB streamed once; floor ~11us at 23.3 TB/s).
//
// CDNA5 path: each lane issues global_load_async_to_lds_b128 (ASYNCcnt, no
// VGPR round-trip, th:TH_LOAD_NT since adj has zero reuse -> don't pollute
// the 192MB L2), then s_wait_asynccnt 0 + barrier.
//
// Selection: each thread caches the argmax of its 16 resident elements in
// registers.  Per round: wave32 __shfl_xor argmax (5 steps) -> 8 per-wave
// candidates in LDS -> thread 0 picks winner (ties -> lower index, matching
// jax.lax.top_k) -> only the owning thread rescans its 16 elements.
// 2 barriers/round instead of ~9, and no full 4096-element rescan.
// ---------------------------------------------------------------------------
__global__ __launch_bounds__(256) void wfm_topk(const float* __restrict__ adj,
                                                int* __restrict__ idx) {
  __shared__ float val[N_];     // 16 KB
  __shared__ float rv8[8];
  __shared__ int   ri8[8];
  __shared__ int   s_win;

  const int row  = blockIdx.x;          // b*N + n
  const int tid  = threadIdx.x;
  const int wave = tid >> 5;
  const int lane = tid & 31;
  const float* a = adj + (size_t)row * N_;

  // ---- async global->LDS stream of the 16KB row (4 x B128 per lane) ------
  {
    const unsigned lds_base = (unsigned)(uintptr_t)(void*)val;  // LDS offset
    for (int j = 0; j < 4; ++j) {
      const int chunk = tid + 256 * j;                 // 16-byte chunks
      unsigned ldsa = lds_base + (unsigned)(chunk * 16);
      unsigned long long ga =
          (unsigned long long)(uintptr_t)(a + (size_t)chunk * 4);
      asm volatile("global_load_async_to_lds_b128 %0, %1, off th:TH_LOAD_NT"
                   :: "v"(ldsa), "v"(ga) : "memory");
    }
    asm volatile("s_wait_asynccnt 0" ::: "memory");
  }
  __syncthreads();

  const float NEG_INF = -__builtin_inff();

  // ---- cached per-thread argmax over my 16 elements (global idx ascending)
  float best = NEG_INF;
  int   bi   = 1 << 30;
  for (int j = 0; j < 4; ++j) {
    const int base = (tid + 256 * j) * 4;
    for (int e = 0; e < 4; ++e) {
      float v = val[base + e];
      if (v > best) { best = v; bi = base + e; }
    }
  }

  for (int r = 0; r < KNN; ++r) {
    // wave32 argmax butterfly (ties -> lower index)
    float bv   = best;
    int   bidx = bi;
    for (int off = 16; off > 0; off >>= 1) {
      float ov = __shfl_xor(bv, off, 32);
      int   oi = __shfl_xor(bidx, off, 32);
      if (ov > bv || (ov == bv && oi < bidx)) { bv = ov; bidx = oi; }
    }
    if (lane == 0) { rv8[wave] = bv; ri8[wave] = bidx; }
    __syncthreads();

    if (tid == 0) {
      float wv = rv8[0]; int wi = ri8[0];
      for (int w = 1; w < 8; ++w) {
        float v2 = rv8[w]; int i2 = ri8[w];
        if (v2 > wv || (v2 == wv && i2 < wi)) { wv = v2; wi = i2; }
      }
      idx[(size_t)row * KNN + r] = wi;
      s_win = wi;
    }
    __syncthreads();

    const int w = s_win;
    if (((w >> 2) & 255) == tid) {        // owner of the extracted element
      val[w] = NEG_INF;
      best = NEG_INF; bi = 1 << 30;       // rescan my 16 elements
      for (int j = 0; j < 4; ++j) {
        const int base = (tid + 256 * j) * 4;
        for (int e = 0; e < 4; ++e) {
          float v = val[base + e];
          if (v > best) { best = v; bi = base + e; }
        }
      }
    }
  }
}

// ---------------------------------------------------------------------------
// Kernel 3: fused gather + log-map + rank-weighted mean (VALU) +
//           [16x16]x[16x32] GEMM via V_WMMA_F32_16X16X4_F32 + exp-map.
//
// 256-thread block = 8 waves; each wave owns one point (b,n).
// Phase 1 (per wave): lane l handles channels (d_a = l>>4, c = l&15) and
//   (d_b = d_a+2, c); loops the 32 sorted neighbors accumulating
//   weighted[d][c] = (1/32) * sum_k q_p_s[k][d][c] * w1n[c][k].
//   cos(theta)=clamped x0 analytically, so only acosf+sinf are needed.
// Phase 2 (waves 0,1): 16 rows = 4 points x D=4; A = weighted (K=C=16 via
//   4 chained k=4 steps), B = w2n[C][M] in two 16-col halves; fp32 WMMA
//   accumulation is numerically equivalent to the fp32 reference.
// Epilogue: per (point, m): mag = ||ws[0..3]||, out = sin(mag)*ws/max(mag,eps),
//   out[d=0] += cos(mag).
// ---------------------------------------------------------------------------
__global__ __launch_bounds__(256) void wfm_main(const float* __restrict__ x,
                                                const int* __restrict__ idx,
                                                const float* __restrict__ w1n,
                                                const float* __restrict__ w2n,
                                                float* __restrict__ out) {
  __shared__ float s_w1n[C_][KNN];   // 2 KB
  __shared__ float s_w2n[C_][M_];    // 2 KB
  __shared__ float s_w[8][D_][C_];   // weighted per point, 2 KB

  const int tid  = threadIdx.x;
  const int wave = tid >> 5;
  const int lane = tid & 31;

  for (int i = tid; i < C_ * KNN; i += 256) s_w1n[i / KNN][i % KNN] = w1n[i];
  for (int i = tid; i < C_ * M_;  i += 256) s_w2n[i / M_][i % M_]   = w2n[i];
  __syncthreads();

  const int gp = blockIdx.x * 8 + wave;   // global point = b*N + n
  const int b  = gp / N_;                 // uniform across the block (N%8==0)

  // ---- Phase 1: weighted[d][c] -------------------------------------------
  {
    const int half = lane >> 4;
    const int c    = lane & 15;
    const int d_a  = half;        // 0 or 1
    const int d_b  = half + 2;    // 2 or 3
    const int*   nbr = idx + (size_t)gp * KNN;
    const float* xb  = x + (size_t)b * N_ * D_ * C_;

    float acc_a = 0.f, acc_b = 0.f;
    for (int k = 0; k < KNN; ++k) {
      const int nk = nbr[k];
      const float* xp = xb + (size_t)nk * D_ * C_;
      float u   = xp[0 * C_ + c];              // axis-0 component
      float xa  = xp[d_a * C_ + c];
      float xbv = xp[d_b * C_ + c];
      u = fminf(1.0f, fmaxf(-1.0f, u));
      float theta = acosf(u);
      float ts    = theta / (sinf(theta) + EPSF);
      float qa    = xa - (d_a == 0 ? u : 0.0f);   // cos(acos(u)) == u
      float wk    = s_w1n[c][k];
      acc_a += qa  * ts * wk;
      acc_b += xbv * ts * wk;                     // d_b in {2,3}: no subtraction
    }
    s_w[wave][d_a][c] = acc_a * (1.0f / (float)KNN);
    s_w[wave][d_b][c] = acc_b * (1.0f / (float)KNN);
  }
  __syncthreads();

  // ---- Phase 2: WMMA GEMM + exp map (wave-uniform branch, EXEC all ones) --
  if (wave < 2) {
    const int p0 = wave * 4;          // 4 points -> 16 rows (point,d)
    const int ll = lane & 15;
    const int h2 = lane >> 4;         // K-half of this lane within the frags
    const int p_local = ll >> 2;      // A-row -> (point, d)
    const int d       = ll & 3;

    v8f acc0 = {};                    // columns m = 0..15
    v8f acc1 = {};                    // columns m = 16..31
    for (int s = 0; s < 4; ++s) {
      const int k0 = 4 * s + 2 * h2;  // lanes 16-31 carry K = 4s+2, 4s+3
      v2f a, b0, b1;
      a[0]  = s_w[p0 + p_local][d][k0 + 0];
      a[1]  = s_w[p0 + p_local][d][k0 + 1];
      b0[0] = s_w2n[k0 + 0][ll];
      b0[1] = s_w2n[k0 + 1][ll];
      b1[0] = s_w2n[k0 + 0][16 + ll];
      b1[1] = s_w2n[k0 + 1][16 + ll];
      acc0 = __builtin_amdgcn_wmma_f32_16x16x4_f32(false, a, false, b0,
                                                   (short)0, acc0, false, false);
      acc1 = __builtin_amdgcn_wmma_f32_16x16x4_f32(false, a, false, b1,
                                                   (short)0, acc1, false, false);
    }

    // D layout: VGPR j = row j (lanes<16) / row j+8 (lanes>=16);
    // row r -> point p0 + (r>>2), d = r&3.
    const int base_gp = blockIdx.x * 8 + p0 + 2 * h2;  // first of this lane-half's 2 points
    for (int h = 0; h < 2; ++h) {
      v8f f = h ? acc1 : acc0;
      const int m = h * 16 + ll;
      for (int pp = 0; pp < 2; ++pp) {
        float f0 = f[pp * 4 + 0], f1 = f[pp * 4 + 1];
        float f2 = f[pp * 4 + 2], f3 = f[pp * 4 + 3];
        float mag   = sqrtf(f0 * f0 + f1 * f1 + f2 * f2 + f3 * f3);
        float scale = sinf(mag) / fmaxf(mag, 1e-12f);
        float o0 = scale * f0 + cosf(mag);
        size_t obase = (size_t)(base_gp + pp) * D_ * M_ + m;
        out[obase + 0 * M_] = o0;
        out[obase + 1 * M_] = scale * f1;
        out[obase + 2 * M_] = scale * f2;
        out[obase + 3 * M_] = scale * f3;
      }
    }
  }
}

// ---------------------------------------------------------------------------
extern "C" void kernel_launch(void* const* d_in, const int* in_sizes, int n_in,
                              void* d_out, int out_size, void* d_ws, size_t ws_size,
                              hipStream_t stream) {
  const float* x   = (const float*)d_in[0];  // [B,N,D,C]
  const float* adj = (const float*)d_in[1];  // [B,N,N]
  const float* w1  = (const float*)d_in[2];  // [C,KNN]
  const float* w2  = (const float*)d_in[3];  // [M,C]
  float* out = (float*)d_out;                // [B,N,D,M]

  // Workspace layout: idx [B*N*KNN] ints, then w1n [C*KNN], w2n [C*M] floats.
  int*   d_idx = (int*)d_ws;
  float* d_w1n = (float*)((char*)d_ws + (size_t)B_ * N_ * KNN * sizeof(int));
  float* d_w2n = d_w1n + C_ * KNN;

  wfm_normalize_weights<<<1, 64, 0, stream>>>(w1, w2, d_w1n, d_w2n);
  wfm_topk<<<B_ * N_, 256, 0, stream>>>(adj, d_idx);
  wfm_main<<<(B_ * N_) / 8, 256, 0, stream>>>(x, d_idx, d_w1n, d_w2n, out);
}